// CharDecodingStep_12627203850551
// MI455X (gfx1250) — compile-verified
//
#include <hip/hip_runtime.h>
#include <math.h>

typedef __attribute__((ext_vector_type(2))) float v2f;
typedef __attribute__((ext_vector_type(8))) float v8f;

#define B_  32
#define S_  2048
#define H_  1024
#define CD_ 256
#define V_  64
#define NS  16              // S-chunks per batch for attention streaming
#define SCHUNK (S_ / NS)    // 128

__device__ __forceinline__ float sigm(float x) { return 1.0f / (1.0f + expf(-x)); }

// ---------------------------------------------------------------------------
// Kernel 1: z = [emb(char) | h_prev] @ [w_ih | w_hh]^T   via V_WMMA_F32_16X16X4_F32
// grid (256, 2): N tiles over 4096, M tiles over 32. One wave per tile.
// A layout (16x4 f32): lanes 0-15 -> M=lane, VGPR0/1 = K0/K1; lanes 16-31 -> K2/K3.
// B layout mirrors with N=lane. C/D: 8 VGPRs, lanes 0-15 N=lane M=r, lanes 16-31 M=8+r.
// ---------------------------------------------------------------------------
__global__ void __launch_bounds__(32)
k_lstm_z(const int* __restrict__ ch, const float* __restrict__ hprev,
         const float* __restrict__ emb, const float* __restrict__ wih,
         const float* __restrict__ whh, float* __restrict__ z)
{
  const int nt = blockIdx.x, mt = blockIdx.y;
  const int lane = threadIdx.x;
  const int ml = lane & 15;
  const int kh = (lane >> 4) << 1;          // 0 or 2
  const int m = mt * 16 + ml;               // batch row 0..31
  const int n = nt * 16 + ml;               // gate column 0..4095

  const float* arow_e = emb   + (size_t)ch[m] * CD_;
  const float* arow_h = hprev + (size_t)m * H_;
  const float* brow_i = wih   + (size_t)n * CD_;
  const float* brow_h = whh   + (size_t)n * H_;

  v8f acc = {0.f,0.f,0.f,0.f,0.f,0.f,0.f,0.f};
#pragma unroll 4
  for (int k0 = 0; k0 < CD_; k0 += 4) {
    v2f a = *(const v2f*)(arow_e + k0 + kh);
    v2f b = *(const v2f*)(brow_i + k0 + kh);
    acc = __builtin_amdgcn_wmma_f32_16x16x4_f32(false, a, false, b, (short)0, acc, false, false);
  }
#pragma unroll 4
  for (int k0 = 0; k0 < H_; k0 += 4) {
    v2f a = *(const v2f*)(arow_h + k0 + kh);
    v2f b = *(const v2f*)(brow_h + k0 + kh);
    acc = __builtin_amdgcn_wmma_f32_16x16x4_f32(false, a, false, b, (short)0, acc, false, false);
  }
  const int cn = nt * 16 + (lane & 15);
  const int mb = mt * 16 + (lane >> 4) * 8;
#pragma unroll
  for (int r = 0; r < 8; ++r)
    z[(size_t)(mb + r) * 4096 + cn] = acc[r];
}

// ---------------------------------------------------------------------------
// Kernel 2: LSTM gates -> c_new, h_new (written straight into d_out regions)
// ---------------------------------------------------------------------------
__global__ void __launch_bounds__(256)
k_lstm_gates(const float* __restrict__ z, const float* __restrict__ cprev,
             const float* __restrict__ bih, const float* __restrict__ bhh,
             float* __restrict__ hnew, float* __restrict__ cnew)
{
  const int idx = blockIdx.x * 256 + threadIdx.x;       // 0..32767
  const int b = idx >> 10, h = idx & 1023;
  const size_t base = (size_t)b * 4096;
  const float zi = z[base + h]            + bih[h]            + bhh[h];
  const float zf = z[base + 1024 + h]     + bih[1024 + h]     + bhh[1024 + h];
  const float zg = z[base + 2048 + h]     + bih[2048 + h]     + bhh[2048 + h];
  const float zo = z[base + 3072 + h]     + bih[3072 + h]     + bhh[3072 + h];
  const float c  = sigm(zf) * cprev[idx] + sigm(zi) * tanhf(zg);
  cnew[idx] = c;
  hnew[idx] = sigm(zo) * tanhf(c);
}

// ---------------------------------------------------------------------------
// Kernel 3: single-pass online-softmax attention over encoder states.
// One wave per (batch, S-chunk). Context accumulator lives in 32 VGPRs/lane.
// Per-row constant (h@w_hid + attn_b) dropped: softmax is shift-invariant and
// masked entries underflow to exactly 0 either way.
// ---------------------------------------------------------------------------
__global__ void __launch_bounds__(32)
k_attn_stream(const float* __restrict__ enc, const float* __restrict__ attn_w,
              const int* __restrict__ lens,
              float* __restrict__ pm, float* __restrict__ pl, float* __restrict__ pctx)
{
  const int w = blockIdx.x;     // chunk id
  const int b = blockIdx.y;     // batch
  const int lane = threadIdx.x;
  const int s0 = w * SCHUNK;
  const int s1 = min(s0 + SCHUNK, lens[b]);

  // preload w_enc slice: lane covers h = j*128 + lane*4 .. +3
  float4 we[8];
#pragma unroll
  for (int j = 0; j < 8; ++j)
    we[j] = *(const float4*)(attn_w + j * 128 + lane * 4);

  float m = -INFINITY, l = 0.0f;
  float4 acc[8];
#pragma unroll
  for (int j = 0; j < 8; ++j) acc[j] = make_float4(0.f, 0.f, 0.f, 0.f);

  const float* base = enc + (size_t)b * S_ * H_;
  for (int s = s0; s < s1; ++s) {
    const float* row = base + (size_t)s * H_;
    if (s + 4 < s1)   // prefetch a future 4KB row (32 lanes x 128B)
      __builtin_prefetch(base + (size_t)(s + 4) * H_ + lane * 32, 0, 0);
    float4 v[8];
#pragma unroll
    for (int j = 0; j < 8; ++j)
      v[j] = *(const float4*)(row + j * 128 + lane * 4);
    float d = 0.f;
#pragma unroll
    for (int j = 0; j < 8; ++j)
      d += v[j].x * we[j].x + v[j].y * we[j].y + v[j].z * we[j].z + v[j].w * we[j].w;
#pragma unroll
    for (int off = 16; off > 0; off >>= 1)
      d += __shfl_xor(d, off, 32);
    const float nm = fmaxf(m, d);
    const float alpha = expf(m - nm);   // exp(-inf)=0 handles first iteration
    const float p = expf(d - nm);
    l = l * alpha + p;
#pragma unroll
    for (int j = 0; j < 8; ++j) {
      acc[j].x = acc[j].x * alpha + p * v[j].x;
      acc[j].y = acc[j].y * alpha + p * v[j].y;
      acc[j].z = acc[j].z * alpha + p * v[j].z;
      acc[j].w = acc[j].w * alpha + p * v[j].w;
    }
    m = nm;
  }
  const int pid = b * NS + w;
  if (lane == 0) { pm[pid] = m; pl[pid] = l; }
  float* dst = pctx + (size_t)pid * H_;
#pragma unroll
  for (int j = 0; j < 8; ++j)
    *(float4*)(dst + j * 128 + lane * 4) = acc[j];
}

// ---------------------------------------------------------------------------
// Kernel 4: merge NS online-softmax partials per batch -> context (b, 1024)
// ---------------------------------------------------------------------------
__global__ void __launch_bounds__(256)
k_attn_reduce(const float* __restrict__ pm, const float* __restrict__ pl,
              const float* __restrict__ pctx, float* __restrict__ ctx)
{
  const int b = blockIdx.x, t = threadIdx.x;
  float gm = -INFINITY;
#pragma unroll
  for (int w = 0; w < NS; ++w) gm = fmaxf(gm, pm[b * NS + w]);
  float sc[NS]; float gl = 0.f;
#pragma unroll
  for (int w = 0; w < NS; ++w) {
    sc[w] = expf(pm[b * NS + w] - gm);     // empty partial: exp(-inf)=0
    gl += sc[w] * pl[b * NS + w];
  }
  const float inv = 1.0f / gl;
  for (int h = t; h < H_; h += 256) {
    float c = 0.f;
#pragma unroll
    for (int w = 0; w < NS; ++w)
      c += sc[w] * pctx[(size_t)(b * NS + w) * H_ + h];
    ctx[(size_t)b * H_ + h] = c * inv;
  }
}

// ---------------------------------------------------------------------------
// Kernel 5: nh = tanh([context | h_new] @ concat_w^T + concat_b)  (WMMA f32)
// ---------------------------------------------------------------------------
__global__ void __launch_bounds__(32)
k_concat_wmma(const float* __restrict__ ctx, const float* __restrict__ hnew,
              const float* __restrict__ cw, const float* __restrict__ cb,
              float* __restrict__ nh)
{
  const int nt = blockIdx.x, mt = blockIdx.y;
  const int lane = threadIdx.x;
  const int ml = lane & 15;
  const int kh = (lane >> 4) << 1;
  const int m = mt * 16 + ml;
  const int n = nt * 16 + ml;

  const float* arow_c = ctx  + (size_t)m * H_;
  const float* arow_h = hnew + (size_t)m * H_;
  const float* brow   = cw   + (size_t)n * (2 * H_);

  v8f acc = {0.f,0.f,0.f,0.f,0.f,0.f,0.f,0.f};
#pragma unroll 4
  for (int k0 = 0; k0 < H_; k0 += 4) {
    v2f a = *(const v2f*)(arow_c + k0 + kh);
    v2f b = *(const v2f*)(brow + k0 + kh);
    acc = __builtin_amdgcn_wmma_f32_16x16x4_f32(false, a, false, b, (short)0, acc, false, false);
  }
#pragma unroll 4
  for (int k0 = 0; k0 < H_; k0 += 4) {
    v2f a = *(const v2f*)(arow_h + k0 + kh);
    v2f b = *(const v2f*)(brow + H_ + k0 + kh);
    acc = __builtin_amdgcn_wmma_f32_16x16x4_f32(false, a, false, b, (short)0, acc, false, false);
  }
  const int cn = nt * 16 + (lane & 15);
  const int mb = mt * 16 + (lane >> 4) * 8;
  const float bias = cb[cn];
#pragma unroll
  for (int r = 0; r < 8; ++r)
    nh[(size_t)(mb + r) * H_ + cn] = tanhf(acc[r] + bias);
}

// ---------------------------------------------------------------------------
// Kernel 6: output_logits = nh @ out_w^T + out_b  (WMMA f32, M=32 N=64 K=1024)
// ---------------------------------------------------------------------------
__global__ void __launch_bounds__(32)
k_out_wmma(const float* __restrict__ nh, const float* __restrict__ ow,
           const float* __restrict__ ob, float* __restrict__ out)
{
  const int nt = blockIdx.x, mt = blockIdx.y;
  const int lane = threadIdx.x;
  const int ml = lane & 15;
  const int kh = (lane >> 4) << 1;
  const int m = mt * 16 + ml;
  const int n = nt * 16 + ml;
  const float* arow = nh + (size_t)m * H_;
  const float* brow = ow + (size_t)n * H_;

  v8f acc = {0.f,0.f,0.f,0.f,0.f,0.f,0.f,0.f};
#pragma unroll 4
  for (int k0 = 0; k0 < H_; k0 += 4) {
    v2f a = *(const v2f*)(arow + k0 + kh);
    v2f b = *(const v2f*)(brow + k0 + kh);
    acc = __builtin_amdgcn_wmma_f32_16x16x4_f32(false, a, false, b, (short)0, acc, false, false);
  }
  const int cn = nt * 16 + (lane & 15);
  const int mb = mt * 16 + (lane >> 4) * 8;
  const float bias = ob[cn];
#pragma unroll
  for (int r = 0; r < 8; ++r)
    out[(size_t)(mb + r) * V_ + cn] = acc[r] + bias;
}

// ---------------------------------------------------------------------------
extern "C" void kernel_launch(void* const* d_in, const int* in_sizes, int n_in,
                              void* d_out, int out_size, void* d_ws, size_t ws_size,
                              hipStream_t stream) {
  (void)in_sizes; (void)n_in; (void)out_size; (void)ws_size;
  const int*   ch     = (const int*)  d_in[0];
  const float* hprev  = (const float*)d_in[1];
  const float* cprev  = (const float*)d_in[2];
  const int*   lens   = (const int*)  d_in[3];
  const float* enc    = (const float*)d_in[4];
  const float* emb    = (const float*)d_in[5];
  const float* wih    = (const float*)d_in[6];
  const float* whh    = (const float*)d_in[7];
  const float* bih    = (const float*)d_in[8];
  const float* bhh    = (const float*)d_in[9];
  const float* attn_w = (const float*)d_in[10];
  // d_in[11] attn_b and the h@w_hid term cancel in softmax (row-constant shift)
  const float* cw     = (const float*)d_in[12];
  const float* cb     = (const float*)d_in[13];
  const float* ow     = (const float*)d_in[14];
  const float* obv    = (const float*)d_in[15];

  float* out    = (float*)d_out;          // [0,2048) logits
  float* hnew   = out + 2048;             // [2048, 34816) h_new
  float* cnew   = out + 2048 + B_ * H_;   // [34816, 67584) c_new

  float* ws   = (float*)d_ws;
  float* z    = ws;                                   // 32*4096
  float* pm   = z + (size_t)B_ * 4096;                // 512
  float* pl   = pm + B_ * NS;                         // 512
  float* pctx = pl + B_ * NS;                         // 512*1024
  float* ctxb = pctx + (size_t)B_ * NS * H_;          // 32*1024
  float* nh   = ctxb + (size_t)B_ * H_;               // 32*1024

  { dim3 g(4096 / 16, B_ / 16);
    k_lstm_z<<<g, 32, 0, stream>>>(ch, hprev, emb, wih, whh, z); }

  k_lstm_gates<<<(B_ * H_) / 256, 256, 0, stream>>>(z, cprev, bih, bhh, hnew, cnew);

  { dim3 g(NS, B_);
    k_attn_stream<<<g, 32, 0, stream>>>(enc, attn_w, lens, pm, pl, pctx); }

  k_attn_reduce<<<B_, 256, 0, stream>>>(pm, pl, pctx, ctxb);

  { dim3 g(H_ / 16, B_ / 16);
    k_concat_wmma<<<g, 32, 0, stream>>>(ctxb, hnew, cw, cb, nh); }

  { dim3 g(V_ / 16, B_ / 16);
    k_out_wmma<<<g, 32, 0, stream>>>(nh, ow, obv, out); }
}